// EncoderRNN_15539191677660
// MI455X (gfx1250) — compile-verified
//
#include <hip/hip_runtime.h>
#include <hip/hip_bf16.h>

typedef __attribute__((ext_vector_type(2))) float v2f;
typedef __attribute__((ext_vector_type(8))) float v8f;

#define E_DIM 300
#define ROWS  900   // 3*E

// Kernel 1: gi = w_ih @ x, gh = w_hh @ h  via fp32 WMMA 16x16x4.
// Grid: 57 blocks (16 rows each), 128 threads (4 waves). Each wave takes a
// strided quarter of the 75 K-chunks; partial column-0 results are reduced in LDS.
__global__ __launch_bounds__(128) void gru_matvec_wmma(
    const int* __restrict__ idx_p,
    const float* __restrict__ emb,
    const float* __restrict__ hid,
    const float* __restrict__ w_ih,
    const float* __restrict__ w_hh,
    float* __restrict__ gi,
    float* __restrict__ gh)
{
    __shared__ float red[2][4][16];

    const int tid  = threadIdx.x;
    const int wave = tid >> 5;
    const int lane = tid & 31;
    const int m0   = blockIdx.x * 16;

    // A-matrix layout (32-bit, 16x4): lanes 0-15 -> K=0,1 ; lanes 16-31 -> K=2,3
    const int koff = (lane >> 4) << 1;      // 0 or 2
    int row = m0 + (lane & 15);
    if (row > ROWS - 1) row = ROWS - 1;     // clamp edge tile (stores are guarded)

    const float* __restrict__ x = emb + (long)idx_p[0] * E_DIM;
    const float* __restrict__ wiRow = w_ih + (long)row * E_DIM;
    const float* __restrict__ whRow = w_hh + (long)row * E_DIM;

    v8f ci = {};
    v8f ch = {};

    // 75 K-chunks of 4, strided across the 4 waves
    for (int t = wave; t < 75; t += 4) {
        const int ks = t * 4 + koff;        // even -> 8B aligned (E_DIM even)
        v2f a_i = *(const v2f*)(wiRow + ks);
        v2f a_h = *(const v2f*)(whRow + ks);
        // B broadcast: every column of B gets the same K-chunk of the vector,
        // using the mirrored K-split (v0: K=0/2, v1: K=1/3 across lane halves).
        v2f b_i = *(const v2f*)(x + ks);
        v2f b_h = *(const v2f*)(hid + ks);

        ci = __builtin_amdgcn_wmma_f32_16x16x4_f32(
                 false, a_i, false, b_i, (short)0, ci, false, false);
        ch = __builtin_amdgcn_wmma_f32_16x16x4_f32(
                 false, a_h, false, b_h, (short)0, ch, false, false);
    }

    // Column 0 of D: lane 0 holds M=0..7 in v0..v7, lane 16 holds M=8..15.
    if ((lane & 15) == 0) {
        const int h8 = (lane >> 4) * 8;
        #pragma unroll
        for (int j = 0; j < 8; ++j) {
            red[0][wave][h8 + j] = ci[j];
            red[1][wave][h8 + j] = ch[j];
        }
    }
    __syncthreads();

    if (tid < 16) {
        const int m = m0 + tid;
        if (m < ROWS) {
            gi[m] = red[0][0][tid] + red[0][1][tid] + red[0][2][tid] + red[0][3][tid];
            gh[m] = red[1][0][tid] + red[1][1][tid] + red[1][2][tid] + red[1][3][tid];
        }
    }
}

// Kernel 2: bias add, gate nonlinearities, h_new, and 3x300 output head.
__global__ __launch_bounds__(320) void gru_gates(
    const float* __restrict__ gi,
    const float* __restrict__ gh,
    const float* __restrict__ b_ih,
    const float* __restrict__ b_hh,
    const float* __restrict__ hid,
    const float* __restrict__ w_out,
    const float* __restrict__ b_out,
    float* __restrict__ out)   // [0..2]=rgb, [3..302]=h_new
{
    __shared__ float hnew[E_DIM];
    const int e = threadIdx.x;

    if (e < E_DIM) {
        const float ir = gi[e]           + b_ih[e];
        const float iz = gi[E_DIM + e]   + b_ih[E_DIM + e];
        const float in = gi[2*E_DIM + e] + b_ih[2*E_DIM + e];
        const float hr = gh[e]           + b_hh[e];
        const float hz = gh[E_DIM + e]   + b_hh[E_DIM + e];
        const float hn = gh[2*E_DIM + e] + b_hh[2*E_DIM + e];

        const float r = 1.0f / (1.0f + expf(-(ir + hr)));
        const float z = 1.0f / (1.0f + expf(-(iz + hz)));
        const float n = tanhf(in + r * hn);
        const float hv = (1.0f - z) * n + z * hid[e];

        hnew[e] = hv;
        out[3 + e] = hv;
    }
    __syncthreads();

    if (e < 3) {
        float acc = b_out[e];
        const float* __restrict__ wrow = w_out + e * E_DIM;
        for (int k = 0; k < E_DIM; ++k) acc += wrow[k] * hnew[k];
        out[e] = 1.0f / (1.0f + expf(-acc));
    }
}

extern "C" void kernel_launch(void* const* d_in, const int* in_sizes, int n_in,
                              void* d_out, int out_size, void* d_ws, size_t ws_size,
                              hipStream_t stream) {
    (void)in_sizes; (void)n_in; (void)out_size; (void)ws_size;

    const int*   idx    = (const int*)  d_in[0];
    const float* hidden = (const float*)d_in[1];
    const float* emb    = (const float*)d_in[2];
    const float* w_ih   = (const float*)d_in[3];
    const float* w_hh   = (const float*)d_in[4];
    const float* b_ih   = (const float*)d_in[5];
    const float* b_hh   = (const float*)d_in[6];
    const float* w_out  = (const float*)d_in[7];
    const float* b_out  = (const float*)d_in[8];
    float*       out    = (float*)d_out;

    float* gi = (float*)d_ws;        // 900 floats
    float* gh = gi + ROWS;           // 900 floats

    gru_matvec_wmma<<<(ROWS + 15) / 16, 128, 0, stream>>>(
        idx, emb, hidden, w_ih, w_hh, gi, gh);
    gru_gates<<<1, 320, 0, stream>>>(
        gi, gh, b_ih, b_hh, hidden, w_out, b_out, out);
}